// GProjRelu_1116691497435
// MI455X (gfx1250) — compile-verified
//
#include <hip/hip_runtime.h>
#include <math.h>

// GProjRelu on MI455X (gfx1250, wave32).
// x:(P,64,16) f32.  out = concat(leaky(x[...,0]),
//   per-segment projected-relu using d = einsum('pik,oi->pok', x_seg, W)).
//
// Per point p, one wave computes dT = x[p]^T (16x64) * W^T (64x64) for BOTH
// W10 and W11 with V_WMMA_F32_16X16X4_F32 (fp32 tensor path = reference
// precision), selecting per D-column in the epilogue.
//
// Data movement (CDNA5-specific): x tiles are DMA'd global->LDS with
// GLOBAL_LOAD_ASYNC_TO_LDS_B128 (ASYNCcnt-tracked), double-buffered so the
// next point's 4KB tile streams in while the current point's 128 WMMAs run.
// ASYNC loads complete in order, so s_wait_asynccnt 8 (<=8 outstanding)
// guarantees the current buffer's 8 B128 transfers have landed.
//
// WMMA fragment layouts (CDNA5 ISA 7.12.2, wave32):
//   A 16x4 f32 : lanes 0-15 -> M=lane, v0=K0,v1=K1 ; lanes 16-31 -> K2,K3
//   B 4x16 f32 : v0: lane<16 -> (K0,N=lane), lane>=16 -> (K2,N=lane-16); v1: K1/K3
//   C/D 16x16  : VGPR r: lane<16 -> (M=r, N=lane), lane>=16 -> (M=r+8, N=lane-16)
// Here M = D-column (0..15), N = output channel, K = input channel.

typedef __attribute__((ext_vector_type(2))) float v2f;
typedef __attribute__((ext_vector_type(8))) float v8f;

#define SLOPE 0.2f
#define EPS2  1e-16f   // EPS^2, inside max() before sqrt

#define WAVES_PER_BLOCK 4
#define THREADS (WAVES_PER_BLOCK * 32)

// Issue 8 async B128 copies: x[p] (4KB) -> LDS buffer at lds_byte.
// GVS addressing: mem = SGPR64(x) + VGPR32(p*4096 + lane*16) + inst_offset.
// On amdgcn a generic pointer to LDS carries the LDS byte offset in its
// low 32 bits (flat-aperture mapping, ISA 10.2), so VDST = (u32)ptr + lane*16.
__device__ __forceinline__ void async_x_prefetch(const float* __restrict__ x,
                                                 int p, const float* lds_buf,
                                                 int lane)
{
    const uint64_t sbase = (uint64_t)(uintptr_t)x;                     // SGPR pair
    const uint32_t voff  = (uint32_t)p * 4096u + (uint32_t)(lane * 16);
    const uint32_t vdst  = (uint32_t)(uintptr_t)lds_buf + (uint32_t)(lane * 16);
    asm volatile(
        "global_load_async_to_lds_b128 %0, %1, %2 offset:0\n\t"
        "global_load_async_to_lds_b128 %0, %1, %2 offset:512\n\t"
        "global_load_async_to_lds_b128 %0, %1, %2 offset:1024\n\t"
        "global_load_async_to_lds_b128 %0, %1, %2 offset:1536\n\t"
        "global_load_async_to_lds_b128 %0, %1, %2 offset:2048\n\t"
        "global_load_async_to_lds_b128 %0, %1, %2 offset:2560\n\t"
        "global_load_async_to_lds_b128 %0, %1, %2 offset:3072\n\t"
        "global_load_async_to_lds_b128 %0, %1, %2 offset:3584"
        :: "v"(vdst), "v"(voff), "s"(sbase) : "memory");
}

__global__ __launch_bounds__(THREADS) void gproj_relu_kernel(
    const float* __restrict__ x,
    const float* __restrict__ W10,
    const float* __restrict__ W11,
    float* __restrict__ out, int P)
{
    __shared__ float Wsh[8192];                       // W10 | W11, 32KB
    __shared__ float Xsh[WAVES_PER_BLOCK][2][1024];   // double-buffered x tiles

    const int tid = threadIdx.x;

    // ---- stage W10/W11 into LDS once per workgroup (coalesced b128) ----
    {
        const float4* s10 = (const float4*)W10;
        const float4* s11 = (const float4*)W11;
        float4* d10 = (float4*)&Wsh[0];
        float4* d11 = (float4*)&Wsh[4096];
        #pragma unroll
        for (int i = tid; i < 1024; i += THREADS) { d10[i] = s10[i]; d11[i] = s11[i]; }
    }
    __syncthreads();

    const int lane = tid & 31;
    const int wib  = tid >> 5;                        // wave in block
    const int wave = blockIdx.x * WAVES_PER_BLOCK + wib;
    const int nwav = gridDim.x * WAVES_PER_BLOCK;

    const int col = lane & 15;                        // D-column group / N-lane
    const int h   = lane >> 4;                        // wave half (0 or 1)

    if (wave >= P) return;

    int buf = 0;
    async_x_prefetch(x, wave, &Xsh[wib][0][0], lane); // prime buffer 0

    for (int p = wave; p < P; p += nwav) {            // uniform per wave
        const int pn = p + nwav;
        if (pn < P) {                                 // overlap next tile's DMA
            async_x_prefetch(x, pn, &Xsh[wib][buf ^ 1][0], lane);
            asm volatile("s_wait_asynccnt 0x8" ::: "memory");
        } else {
            asm volatile("s_wait_asynccnt 0x0" ::: "memory");
        }
        const float* xs = &Xsh[wib][buf][0];

        // ---- build the 16 A fragments of x^T (kept in registers) ----
        v2f a[16];
        #pragma unroll
        for (int q = 0; q < 16; ++q) {                // K chunk: channels 4q..4q+3
            const int i0 = 4 * q + 2 * h;
            v2f av; av.x = xs[i0 * 16 + col];
                    av.y = xs[(i0 + 1) * 16 + col];
            a[q] = av;
        }

        // ---- 4 output-channel tiles; per tile: 2 x 16 chained WMMAs ----
        #pragma unroll
        for (int t = 0; t < 4; ++t) {
            const int c = 16 * t + col;               // this lane's out-channel
            v8f d10 = {}; v8f d11 = {};
            #pragma unroll
            for (int q = 0; q < 16; ++q) {            // B = W10^T frag: contiguous pair
                const int i0 = 4 * q + 2 * h;
                v2f b = *(const v2f*)&Wsh[c * 64 + i0];
                d10 = __builtin_amdgcn_wmma_f32_16x16x4_f32(
                          false, a[q], false, b, (short)0, d10, false, false);
            }
            #pragma unroll
            for (int q = 0; q < 16; ++q) {            // B = W11^T frag
                const int i0 = 4 * q + 2 * h;
                v2f b = *(const v2f*)&Wsh[4096 + c * 64 + i0];
                d11 = __builtin_amdgcn_wmma_f32_16x16x4_f32(
                          false, a[q], false, b, (short)0, d11, false, false);
            }

            // ---- epilogue: lane holds D-columns 8h..8h+7 of channel c ----
            float xr[8], g10[8], g11[8];
            #pragma unroll
            for (int r = 0; r < 8; ++r) {
                xr[r]  = xs[c * 16 + 8 * h + r];
                g10[r] = d10[r];
                g11[r] = d11[r];
            }

            float dotA = 0.f, nA = 0.f, dotB = 0.f, nB = 0.f;
            float pC = 0.f, pnC = 0.f;
            if (h == 0) {
                dotA = g10[1]*xr[1] + g10[2]*xr[2] + g10[3]*xr[3];
                nA   = g10[1]*g10[1] + g10[2]*g10[2] + g10[3]*g10[3];
                dotB = g10[4]*xr[4] + g10[5]*xr[5] + g10[6]*xr[6];
                nB   = g10[4]*g10[4] + g10[5]*g10[5] + g10[6]*g10[6];
                pC   = g11[7]*xr[7];                  // column 7 partial
                pnC  = g11[7]*g11[7];
            } else {
                #pragma unroll
                for (int r = 0; r < 8; ++r) {         // columns 8..15 partials
                    pC  += g11[r]*xr[r];
                    pnC += g11[r]*g11[r];
                }
            }
            // segment C spans both wave halves: exchange lane <-> lane+16
            const float dotC = pC  + __shfl_xor(pC,  16, 32);
            const float nC   = pnC + __shfl_xor(pnC, 16, 32);

            const float sA = dotA / sqrtf(fmaxf(nA, EPS2));
            const float sB = dotB / sqrtf(fmaxf(nB, EPS2));
            const float sC = dotC / sqrtf(fmaxf(nC, EPS2));

            float o[8];
            if (h == 0) {
                o[0] = (xr[0] >= 0.f) ? xr[0] : SLOPE * xr[0];   // leaky col 0
                #pragma unroll
                for (int r = 1; r <= 3; ++r) {
                    float relu = (dotA >= 0.f) ? xr[r] : (xr[r] - g10[r] * sA);
                    o[r] = SLOPE * xr[r] + (1.f - SLOPE) * relu;
                }
                #pragma unroll
                for (int r = 4; r <= 6; ++r) {
                    float relu = (dotB >= 0.f) ? xr[r] : (xr[r] - g10[r] * sB);
                    o[r] = SLOPE * xr[r] + (1.f - SLOPE) * relu;
                }
                {
                    float relu = (dotC >= 0.f) ? xr[7] : (xr[7] - g11[7] * sC);
                    o[7] = SLOPE * xr[7] + (1.f - SLOPE) * relu;
                }
            } else {
                #pragma unroll
                for (int r = 0; r < 8; ++r) {
                    float relu = (dotC >= 0.f) ? xr[r] : (xr[r] - g11[r] * sC);
                    o[r] = SLOPE * xr[r] + (1.f - SLOPE) * relu;
                }
            }

            float4* og = (float4*)(out + (size_t)p * 1024 + c * 16 + 8 * h);
            og[0] = make_float4(o[0], o[1], o[2], o[3]);
            og[1] = make_float4(o[4], o[5], o[6], o[7]);
        }
        buf ^= 1;
    }
}

extern "C" void kernel_launch(void* const* d_in, const int* in_sizes, int n_in,
                              void* d_out, int out_size, void* d_ws, size_t ws_size,
                              hipStream_t stream) {
    const float* x   = (const float*)d_in[0];
    const float* W10 = (const float*)d_in[1];
    const float* W11 = (const float*)d_in[2];
    float* out = (float*)d_out;
    const int P = in_sizes[0] / (64 * 16);            // 65536

    const dim3 grid(2048), block(THREADS);            // 8192 waves, 8 pts each
    hipLaunchKernelGGL(gproj_relu_kernel, grid, block, 0, stream,
                       x, W10, W11, out, P);
}